// STFTLoss_78649441124700
// MI455X (gfx1250) — compile-verified
//
#include <hip/hip_runtime.h>
#include <hip/hip_bf16.h>
#include <math.h>

typedef __attribute__((ext_vector_type(2))) float v2f;
typedef __attribute__((ext_vector_type(8))) float v8f;
typedef __attribute__((ext_vector_type(2))) int   v2i;

#define NFFT      1024
#define FRAME_LEN 256
#define LP        384            // (NFFT - FRAME_LEN)/2
#define STEP      128
#define NBINS     513
#define NPAD      544            // 34 tiles of 16 bins -> 17 tile-pairs
#define FRAMES    4096
#define T_LEN     524288
#define TP_LEN    (T_LEN + NFFT)
#define EPSV      1e-8f
#define NBT2      (NPAD / 32)    // 17 bin-tile-pairs
#define NFT2      (FRAMES / 32)  // 128 frame-tile-pairs
#define WAVES_PER_BLOCK 4
#define NPART     (NFT2 * NBT2)  // 2176

// ---- workspace layout (float offsets) ----
#define WS_WRE   0
#define WS_WIM   (NPAD * FRAME_LEN)
#define WS_XP    (2 * NPAD * FRAME_LEN)
#define WS_YP    (WS_XP + TP_LEN)
#define WS_PART  (WS_YP + TP_LEN)

#if __has_builtin(__builtin_amdgcn_global_load_async_to_lds_b64) && \
    __has_builtin(__builtin_amdgcn_s_wait_asynccnt)
#define USE_ASYNC_LDS 1
#endif

// Copy 8 bytes global -> LDS (per-lane addresses). Async path uses the CDNA5
// GLOBAL_LOAD_ASYNC_TO_LDS_B64 op (ASYNCcnt-tracked); fallback is a plain copy.
__device__ __forceinline__ void stage_b64(const float* g, float* l) {
#ifdef USE_ASYNC_LDS
    typedef __attribute__((address_space(1))) v2i* gp_t;  // global ptr to int2
    typedef __attribute__((address_space(3))) v2i* lp_t;  // LDS ptr to int2
    __builtin_amdgcn_global_load_async_to_lds_b64((gp_t)g, (lp_t)l, 0, 0);
#else
    *(v2f*)l = *(const v2f*)g;
#endif
}

// W stored transposed: W[n_bin][t_tap] so staging reads contiguous float2s.
__global__ void k_init_weights(float* __restrict__ wre, float* __restrict__ wim) {
    int idx = blockIdx.x * blockDim.x + threadIdx.x;
    if (idx >= NPAD * FRAME_LEN) return;
    int n = idx / FRAME_LEN;   // bin (0..543; >=513 zero pad)
    int t = idx % FRAME_LEN;   // tap (0..255)
    float re = 0.f, im = 0.f;
    if (n < NBINS) {
        const float PI = 3.14159265358979323846f;
        float win  = 0.5f - 0.5f * cosf(2.f * PI * (float)t / (float)FRAME_LEN);
        int   imod = ((t + LP) * n) % NFFT;                 // exact integer phase
        float th   = (-2.f * PI / (float)NFFT) * (float)imod;
        re = cosf(th) * win;
        im = sinf(th) * win;
    }
    wre[idx] = re;
    wim[idx] = im;
}

// End-zero-pad row 7 of each signal so A-fragment loads are unconditional.
__global__ void k_pad(const float* __restrict__ x, const float* __restrict__ y,
                      float* __restrict__ xp, float* __restrict__ yp) {
    int i = blockIdx.x * blockDim.x + threadIdx.x;
    if (i >= TP_LEN) return;
    xp[i] = (i < T_LEN) ? x[i] : 0.f;
    yp[i] = (i < T_LEN) ? y[i] : 0.f;
}

// Block = 4 waves. blockIdx.y selects a bin-tile-PAIR (32 bins); the block
// async-stages its (re,im) weights into LDS pre-swizzled into B-fragment
// layout. Each wave owns a frame-tile-PAIR (32 frames) and computes a
// 2(frame-tile) x 2(bin-tile) register tile: 16 f32 WMMA accumulators,
// 16 v_wmma_f32_16x16x4_f32 per 4 global_load_b64 + 4 ds_load_b64.
__global__ void __launch_bounds__(32 * WAVES_PER_BLOCK)
k_stft_loss(const float* __restrict__ xp, const float* __restrict__ yp,
            const float* __restrict__ wre, const float* __restrict__ wim,
            float* __restrict__ part) {
    // lds_w[(arr*2+bt)*4096 + ks*64 + lane*2] : exactly 64KB, conflict-free reads
    __shared__ float lds_w[4 * 64 * 64];

    const int btpair = blockIdx.y;           // 0..16
    const int wv     = threadIdx.x >> 5;     // 0..3
    const int lane   = threadIdx.x & 31;
    const int m      = lane & 15;
    const int half   = lane >> 4;            // K-pair select: {0,1} vs {2,3}

    // ---- stage weights: wave wv stages (arr = wv>>1 : re/im, bt = wv&1) ----
    {
        const int arr = wv >> 1, bt = wv & 1;
        const float* wsrc = arr ? wim : wre;
        const int bin = btpair * 32 + bt * 16 + m;
        const float* g = wsrc + (size_t)bin * FRAME_LEN + 2 * half;
        float*       l = lds_w + (arr * 2 + bt) * 4096 + lane * 2;
#pragma unroll 8
        for (int ks = 0; ks < 64; ++ks)
            stage_b64(g + ks * 4, l + ks * 64);
    }
#ifdef USE_ASYNC_LDS
    __builtin_amdgcn_s_wait_asynccnt(0);
#endif
    __syncthreads();

    // ---- GEMM: frame-tile-pair fp, bin-tiles {2*btpair, 2*btpair+1} ----
    const int fp = blockIdx.x * WAVES_PER_BLOCK + wv;   // 0..127
    const float* ax0 = xp + (size_t)(fp * 32 + m)      * STEP + LP + 2 * half;
    const float* ax1 = xp + (size_t)(fp * 32 + 16 + m) * STEP + LP + 2 * half;
    const float* ay0 = yp + (size_t)(fp * 32 + m)      * STEP + LP + 2 * half;
    const float* ay1 = yp + (size_t)(fp * 32 + 16 + m) * STEP + LP + 2 * half;
    const float* lr0 = lds_w + 0 * 4096 + lane * 2;   // re, bt0
    const float* lr1 = lds_w + 1 * 4096 + lane * 2;   // re, bt1
    const float* li0 = lds_w + 2 * 4096 + lane * 2;   // im, bt0
    const float* li1 = lds_w + 3 * 4096 + lane * 2;   // im, bt1

    v8f xr00 = {}, xr01 = {}, xr10 = {}, xr11 = {};
    v8f xi00 = {}, xi01 = {}, xi10 = {}, xi11 = {};
    v8f yr00 = {}, yr01 = {}, yr10 = {}, yr11 = {};
    v8f yi00 = {}, yi01 = {}, yi10 = {}, yi11 = {};

#pragma unroll 4
    for (int ks = 0; ks < 64; ++ks) {
        v2f a_x0 = *(const v2f*)(ax0 + ks * 4);
        v2f a_x1 = *(const v2f*)(ax1 + ks * 4);
        v2f a_y0 = *(const v2f*)(ay0 + ks * 4);
        v2f a_y1 = *(const v2f*)(ay1 + ks * 4);
        v2f b_r0 = *(const v2f*)(lr0 + ks * 64);
        v2f b_r1 = *(const v2f*)(lr1 + ks * 64);
        v2f b_i0 = *(const v2f*)(li0 + ks * 64);
        v2f b_i1 = *(const v2f*)(li1 + ks * 64);
        xr00 = __builtin_amdgcn_wmma_f32_16x16x4_f32(false, a_x0, false, b_r0, (short)0, xr00, false, false);
        xr01 = __builtin_amdgcn_wmma_f32_16x16x4_f32(false, a_x0, false, b_r1, (short)0, xr01, false, false);
        xi00 = __builtin_amdgcn_wmma_f32_16x16x4_f32(false, a_x0, false, b_i0, (short)0, xi00, false, false);
        xi01 = __builtin_amdgcn_wmma_f32_16x16x4_f32(false, a_x0, false, b_i1, (short)0, xi01, false, false);
        xr10 = __builtin_amdgcn_wmma_f32_16x16x4_f32(false, a_x1, false, b_r0, (short)0, xr10, false, false);
        xr11 = __builtin_amdgcn_wmma_f32_16x16x4_f32(false, a_x1, false, b_r1, (short)0, xr11, false, false);
        xi10 = __builtin_amdgcn_wmma_f32_16x16x4_f32(false, a_x1, false, b_i0, (short)0, xi10, false, false);
        xi11 = __builtin_amdgcn_wmma_f32_16x16x4_f32(false, a_x1, false, b_i1, (short)0, xi11, false, false);
        yr00 = __builtin_amdgcn_wmma_f32_16x16x4_f32(false, a_y0, false, b_r0, (short)0, yr00, false, false);
        yr01 = __builtin_amdgcn_wmma_f32_16x16x4_f32(false, a_y0, false, b_r1, (short)0, yr01, false, false);
        yi00 = __builtin_amdgcn_wmma_f32_16x16x4_f32(false, a_y0, false, b_i0, (short)0, yi00, false, false);
        yi01 = __builtin_amdgcn_wmma_f32_16x16x4_f32(false, a_y0, false, b_i1, (short)0, yi01, false, false);
        yr10 = __builtin_amdgcn_wmma_f32_16x16x4_f32(false, a_y1, false, b_r0, (short)0, yr10, false, false);
        yr11 = __builtin_amdgcn_wmma_f32_16x16x4_f32(false, a_y1, false, b_r1, (short)0, yr11, false, false);
        yi10 = __builtin_amdgcn_wmma_f32_16x16x4_f32(false, a_y1, false, b_i0, (short)0, yi10, false, false);
        yi11 = __builtin_amdgcn_wmma_f32_16x16x4_f32(false, a_y1, false, b_i1, (short)0, yi11, false, false);
    }

    // ---- epilogue: magnitude + loss contribution, mask bins >= 513 ----
    float acc = 0.f;
    const int nb0 = btpair * 32 + m;
    const int nb1 = nb0 + 16;
#pragma unroll
    for (int v = 0; v < 8; ++v) {
        if (nb0 < NBINS) {
            {   // ft0, bt0
                float xm = sqrtf(fmaxf(xr00[v]*xr00[v] + xi00[v]*xi00[v], EPSV));
                float ym = sqrtf(fmaxf(yr00[v]*yr00[v] + yi00[v]*yi00[v], EPSV));
                acc += fabsf(logf(xm) - logf(ym)) + fabsf(xm - ym);
            }
            {   // ft1, bt0
                float xm = sqrtf(fmaxf(xr10[v]*xr10[v] + xi10[v]*xi10[v], EPSV));
                float ym = sqrtf(fmaxf(yr10[v]*yr10[v] + yi10[v]*yi10[v], EPSV));
                acc += fabsf(logf(xm) - logf(ym)) + fabsf(xm - ym);
            }
        }
        if (nb1 < NBINS) {
            {   // ft0, bt1
                float xm = sqrtf(fmaxf(xr01[v]*xr01[v] + xi01[v]*xi01[v], EPSV));
                float ym = sqrtf(fmaxf(yr01[v]*yr01[v] + yi01[v]*yi01[v], EPSV));
                acc += fabsf(logf(xm) - logf(ym)) + fabsf(xm - ym);
            }
            {   // ft1, bt1
                float xm = sqrtf(fmaxf(xr11[v]*xr11[v] + xi11[v]*xi11[v], EPSV));
                float ym = sqrtf(fmaxf(yr11[v]*yr11[v] + yi11[v]*yi11[v], EPSV));
                acc += fabsf(logf(xm) - logf(ym)) + fabsf(xm - ym);
            }
        }
    }
    // deterministic wave32 reduction
    for (int off = 16; off > 0; off >>= 1)
        acc += __shfl_xor(acc, off, 32);
    if (lane == 0)
        part[fp * NBT2 + btpair] = acc;
}

// Fixed-order final reduction (deterministic; no float atomics).
__global__ void k_reduce(const float* __restrict__ part, float* __restrict__ out) {
    __shared__ float s[256];
    int tid = threadIdx.x;
    float v = 0.f;
    for (int i = tid; i < NPART; i += 256) v += part[i];
    s[tid] = v;
    __syncthreads();
    for (int stride = 128; stride > 0; stride >>= 1) {
        if (tid < stride) s[tid] += s[tid + stride];
        __syncthreads();
    }
    if (tid == 0) out[0] = s[0] / (float)(FRAMES * NBINS);
}

extern "C" void kernel_launch(void* const* d_in, const int* in_sizes, int n_in,
                              void* d_out, int out_size, void* d_ws, size_t ws_size,
                              hipStream_t stream) {
    (void)in_sizes; (void)n_in; (void)out_size; (void)ws_size;
    // Only the LAST batch row (index 7) is used by the reference.
    const float* x = (const float*)d_in[0] + (size_t)7 * T_LEN;
    const float* y = (const float*)d_in[1] + (size_t)7 * T_LEN;

    float* ws   = (float*)d_ws;
    float* wre  = ws + WS_WRE;
    float* wim  = ws + WS_WIM;
    float* xp   = ws + WS_XP;
    float* yp   = ws + WS_YP;
    float* part = ws + WS_PART;

    k_init_weights<<<(NPAD * FRAME_LEN + 255) / 256, 256, 0, stream>>>(wre, wim);
    k_pad<<<(TP_LEN + 255) / 256, 256, 0, stream>>>(x, y, xp, yp);

    dim3 grid(NFT2 / WAVES_PER_BLOCK, NBT2);   // 32 x 17 blocks, 4 waves each
    k_stft_loss<<<grid, 32 * WAVES_PER_BLOCK, 0, stream>>>(xp, yp, wre, wim, part);

    k_reduce<<<1, 256, 0, stream>>>(part, (float*)d_out);
}